// OFA_78357383348248
// MI455X (gfx1250) — compile-verified
//
#include <hip/hip_runtime.h>
#include <math.h>

// ---------------------------------------------------------------------------
// Problem constants (fixed by the reference)
// ---------------------------------------------------------------------------
#define G_NUM      1024          // graphs
#define NPG        256           // nodes per graph
#define INF        768           // IN_FEATS
#define HID        128           // HIDDEN == OUT_FEATS
#define NCLS       2

#define C1_NORM    0.04393747805f   // 1/sqrt(2*259)
#define INV259     (1.0f/259.0f)

typedef __attribute__((ext_vector_type(2))) float v2f;
typedef __attribute__((ext_vector_type(8))) float v8f;

// ---------------------------------------------------------------------------
// K1: per-graph column sum of x.  S[g][c] = sum_{r=0..255} x[g*256+r][c]
// grid (1024, 3), block 256 (64 float4 lanes x 4 row-strides).
// Streams the full 768 MB of x with coalesced float4 loads -> HBM bound.
// ---------------------------------------------------------------------------
__global__ void k1_graph_colsum(const float* __restrict__ x, float* __restrict__ S) {
    const int g  = blockIdx.x;        // graph
    const int cc = blockIdx.y;        // 256-float column chunk (0..2)
    const int tx = threadIdx.x & 63;  // float4 column within chunk
    const int ty = threadIdx.x >> 6;  // row stride phase (0..3)

    const float4* xp = (const float4*)x + (size_t)g * NPG * (INF / 4)
                       + (size_t)cc * 64 + tx;
    float4 acc = make_float4(0.f, 0.f, 0.f, 0.f);
#pragma unroll 4
    for (int r = ty; r < NPG; r += 4) {
        float4 v = xp[(size_t)r * (INF / 4)];
        acc.x += v.x; acc.y += v.y; acc.z += v.z; acc.w += v.w;
    }

    __shared__ float4 red[256];
    red[threadIdx.x] = acc;
    __syncthreads();
    if (ty == 0) {
        float4 a = red[tx], b = red[64 + tx], c = red[128 + tx], d = red[192 + tx];
        float4 s = make_float4(a.x + b.x + c.x + d.x,
                               a.y + b.y + c.y + d.y,
                               a.z + b.z + c.z + d.z,
                               a.w + b.w + c.w + d.w);
        ((float4*)S)[(size_t)g * (INF / 4) + cc * 64 + tx] = s;
    }
}

// ---------------------------------------------------------------------------
// K2: T = S @ W1  (M=1024, K=768, N=128), full f32 via V_WMMA_F32_16X16X4_F32.
// One wave per 16x16 tile of T.  grid 64 blocks x 256 threads (8 waves):
// blockIdx.x -> M tile, wave id -> N tile.  K stepped by 4.
//
// ISA lane layouts (CDNA5 ISA):
//   A 16x4 f32 : lanes 0-15 M=0..15 {K=0,K=1}; lanes 16-31 {K=2,K=3}
//   B 4x16 f32 : mirrored (rows K striped the same way, N = lane&15)
//   C/D 16x16  : VGPR v -> M=v (lanes 0-15) / M=v+8 (lanes 16-31), N = lane&15
// ---------------------------------------------------------------------------
__global__ void k2_gemm_T(const float* __restrict__ S,
                          const float* __restrict__ W1,
                          float* __restrict__ T) {
    const int wave = threadIdx.x >> 5;      // 0..7 -> N tile
    const int lane = threadIdx.x & 31;
    const int m0   = blockIdx.x * 16;
    const int n0   = wave * 16;
    const int ml   = lane & 15;
    const int kh   = (lane >> 4) << 1;      // +0 (lanes 0-15) or +2 (lanes 16-31)

    const float* Srow = S + (size_t)(m0 + ml) * INF;
    const float* Wcol = W1 + n0 + ml;

    v8f acc = {0.f, 0.f, 0.f, 0.f, 0.f, 0.f, 0.f, 0.f};
#pragma unroll 8
    for (int k0 = 0; k0 < INF; k0 += 4) {
        const int kk = k0 + kh;
        v2f a, b;
        a.x = Srow[kk];
        a.y = Srow[kk + 1];
        b.x = Wcol[(size_t)kk * HID];
        b.y = Wcol[(size_t)(kk + 1) * HID];
        acc = __builtin_amdgcn_wmma_f32_16x16x4_f32(
            /*neg_a=*/false, a, /*neg_b=*/false, b,
            /*c_mod=*/(short)0, acc, /*reuse_a=*/false, /*reuse_b=*/false);
    }

    const int rowoff = (lane < 16) ? 0 : 8;
    const int col    = n0 + ml;
#pragma unroll
    for (int v = 0; v < 8; ++v)
        T[(size_t)(m0 + v + rowoff) * HID + col] = acc[v];
}

// ---------------------------------------------------------------------------
// K3: O(1) precompute (graph-independent vectors).  1 block x 128 threads.
//   hp     = prompt @ W1
//   hcsum  = (cls0 + cls1) @ W1
//   h2c[c] = relu(c1*hp + 0.5*hc_c + b1) @ W2
// ---------------------------------------------------------------------------
__global__ void k3_precompute(const float* __restrict__ prompt,
                              const float* __restrict__ cls,
                              const float* __restrict__ W1,
                              const float* __restrict__ b1,
                              const float* __restrict__ W2,
                              float* __restrict__ hp,
                              float* __restrict__ hcsum,
                              float* __restrict__ h2c) {
    const int j = threadIdx.x;  // 0..127
    float sp = 0.f, sc0 = 0.f, sc1 = 0.f;
    for (int k = 0; k < INF; ++k) {
        const float w = W1[(size_t)k * HID + j];
        sp  += prompt[k] * w;
        sc0 += cls[k] * w;
        sc1 += cls[INF + k] * w;
    }
    hp[j]    = sp;
    hcsum[j] = sc0 + sc1;

    const float bb = b1[j];
    __shared__ float r0[HID], r1[HID];
    r0[j] = fmaxf(C1_NORM * sp + 0.5f * sc0 + bb, 0.f);
    r1[j] = fmaxf(C1_NORM * sp + 0.5f * sc1 + bb, 0.f);
    __syncthreads();

    float t0 = 0.f, t1 = 0.f;
    for (int k = 0; k < HID; ++k) {
        const float w = W2[(size_t)k * HID + j];
        t0 += r0[k] * w;
        t1 += r1[k] * w;
    }
    h2c[j]       = t0;
    h2c[HID + j] = t1;
}

// ---------------------------------------------------------------------------
// K4: per-graph finish.  grid 1024 x 128 threads.
//   r     = relu(c1*(T[g] + hcsum) + hp/259 + b1)
//   h2p   = r @ W2
//   o_c   = c1*h2p + 0.5*h2c[c] + b2
//   out   = log_softmax(o_c @ Wfc + bfc)
// ---------------------------------------------------------------------------
__global__ void k4_finish(const float* __restrict__ T,
                          const float* __restrict__ hp,
                          const float* __restrict__ hcsum,
                          const float* __restrict__ h2c,
                          const float* __restrict__ b1,
                          const float* __restrict__ W2,
                          const float* __restrict__ b2,
                          const float* __restrict__ Wfc,
                          const float* __restrict__ bfc,
                          float* __restrict__ out) {
    const int g = blockIdx.x;
    const int j = threadIdx.x;

    __shared__ float r[HID];
    __shared__ float o[2][HID];

    const float h1 = C1_NORM * (T[(size_t)g * HID + j] + hcsum[j])
                     + INV259 * hp[j] + b1[j];
    r[j] = fmaxf(h1, 0.f);
    __syncthreads();

    float t = 0.f;
    for (int k = 0; k < HID; ++k)
        t += r[k] * W2[(size_t)k * HID + j];

    const float bb = b2[j];
    o[0][j] = C1_NORM * t + 0.5f * h2c[j] + bb;
    o[1][j] = C1_NORM * t + 0.5f * h2c[HID + j] + bb;
    __syncthreads();

    if (j < NCLS) {  // thread j handles class node j of this graph
        float l0 = bfc[0], l1 = bfc[1];
        for (int k = 0; k < HID; ++k) {
            const float v = o[j][k];
            l0 += v * Wfc[k * NCLS + 0];
            l1 += v * Wfc[k * NCLS + 1];
        }
        const float m   = fmaxf(l0, l1);
        const float lse = m + logf(expf(l0 - m) + expf(l1 - m));
        float* po = out + ((size_t)g * NCLS + j) * NCLS;
        po[0] = l0 - lse;
        po[1] = l1 - lse;
    }
}

// ---------------------------------------------------------------------------
// launch
// ---------------------------------------------------------------------------
extern "C" void kernel_launch(void* const* d_in, const int* in_sizes, int n_in,
                              void* d_out, int out_size, void* d_ws, size_t ws_size,
                              hipStream_t stream) {
    const float* x      = (const float*)d_in[0];
    const float* prompt = (const float*)d_in[1];
    const float* cls    = (const float*)d_in[2];
    const float* W1     = (const float*)d_in[3];
    const float* b1     = (const float*)d_in[4];
    const float* W2     = (const float*)d_in[5];
    const float* b2     = (const float*)d_in[6];
    const float* Wfc    = (const float*)d_in[7];
    const float* bfc    = (const float*)d_in[8];
    float* out          = (float*)d_out;

    // workspace layout (floats)
    float* ws    = (float*)d_ws;
    float* S     = ws;                                  // 1024*768
    float* T     = S + (size_t)G_NUM * INF;             // 1024*128
    float* hp    = T + (size_t)G_NUM * HID;             // 128
    float* hcsum = hp + HID;                            // 128
    float* h2c   = hcsum + HID;                         // 256

    // K1: per-graph feature sums (HBM-bound streaming of x)
    k1_graph_colsum<<<dim3(G_NUM, 3), 256, 0, stream>>>(x, S);

    // K2: T = S @ W1 via f32 WMMA (64 M-tiles x 8 N-tiles, 1 wave/tile)
    k2_gemm_T<<<dim3(G_NUM / 16), 256, 0, stream>>>(S, W1, T);

    // K3: graph-independent vectors
    k3_precompute<<<1, HID, 0, stream>>>(prompt, cls, W1, b1, W2, hp, hcsum, h2c);

    // K4: per-graph matvec + head + log_softmax
    k4_finish<<<dim3(G_NUM), HID, 0, stream>>>(T, hp, hcsum, h2c, b1, W2, b2,
                                               Wfc, bfc, out);
}